// ErrorInterpolate_28767690948642
// MI455X (gfx1250) — compile-verified
//
#include <hip/hip_runtime.h>

typedef __attribute__((ext_vector_type(2))) float v2f;
typedef __attribute__((ext_vector_type(8))) float v8f;

#define KNN 32
#define FDIM 128
#define FINE_PER_WAVE 16
#define FINE_PER_BLOCK 64
#define BLOCK 128

// One wave owns 16 fine points. Per chunk of 16 coarse points, a single
// V_WMMA_F32_16X16X4_F32 produces the full 16x16 squared-distance tile:
//   A[m][:] = [lx, ly, lz, |l|^2]    (coarse chunk, M x K)
//   B[:][n] = [-2hx, -2hy, -2hz, 1]  (fine tile,   K x N)
//   C       = broadcast(|h|^2)
//   D[m][n] = |h|^2 + |l|^2 - 2 dot(l,h)  == d2
// C/D layout: lane L holds fine column n = L%16; VGPR j holds coarse row
// m = j + 8*(L/16). So lane L keeps a sorted top-32 over HALF the coarse
// candidates of fine point (L%16); lanes f and f+16 merge at the end.
// The pos_l fetch for chunk c+1 is software-pipelined (issued before the
// divergent insert region of chunk c) so the global-load latency overlaps
// the WMMA + top-K update instead of stalling at s_wait_loadcnt each iter.
// NOTE: single insert path only — duplicating it (full/tail variants) makes
// the register allocator shuffle the 64-reg top-K arrays with v_mov walls.
__global__ __launch_bounds__(BLOCK) void knn_interp_wmma_kernel(
    const float* __restrict__ x, const float* __restrict__ pos_l,
    const float* __restrict__ pos_h, float* __restrict__ out,
    int N_l, int N_h)
{
    __shared__ float s_ld[BLOCK * KNN];            // per-lane candidate d2 lists
    __shared__ int   s_li[BLOCK * KNN];            // per-lane candidate idx lists
    __shared__ float s_w [FINE_PER_BLOCK * KNN];   // merged raw weights
    __shared__ int   s_wi[FINE_PER_BLOCK * KNN];   // merged indices
    __shared__ float s_den[FINE_PER_BLOCK];        // weight sums

    const int tid  = threadIdx.x;
    const int lane = tid & 31;
    const int wave = tid >> 5;
    const int half = lane >> 4;      // which K-pair / which coarse row half
    const int n    = lane & 15;      // fine column within the wave's tile

    const int fineBase = blockIdx.x * FINE_PER_BLOCK;
    const int gf = fineBase + wave * FINE_PER_WAVE + n;

    // ---- Fine-point operand (B) and C initializer ----
    float hx = 0.f, hy = 0.f, hz = 0.f;
    if (gf < N_h) {
        hx = pos_h[3 * gf + 0];
        hy = pos_h[3 * gf + 1];
        hz = pos_h[3 * gf + 2];
    }
    const float sqh = hx * hx + hy * hy + hz * hz;

    v2f B;
    B.x = half ? (-2.0f * hz) : (-2.0f * hx);   // K=2 : K=0
    B.y = half ? 1.0f         : (-2.0f * hy);   // K=3 : K=1
    const v8f Cinit = {sqh, sqh, sqh, sqh, sqh, sqh, sqh, sqh};

    // ---- Register-resident sorted top-K (ascending, worst at [KNN-1]) ----
    float bd[KNN];
    int   bi[KNN];
#pragma unroll
    for (int s = 0; s < KNN; ++s) { bd[s] = 3.4e38f; bi[s] = 0; }

    const int chunks = (N_l + 15) >> 4;

    // Prime the pipeline: load chunk 0's coarse point for this lane.
    int mc0 = n < N_l ? n : (N_l - 1);
    float lx = pos_l[3 * mc0 + 0];
    float ly = pos_l[3 * mc0 + 1];
    float lz = pos_l[3 * mc0 + 2];

    for (int c = 0; c < chunks; ++c) {
        // ---- Build A for chunk c from the prefetched registers ----
        float sql = lx * lx + ly * ly + lz * lz;
        v2f A;
        A.x = half ? lz  : lx;                   // K=2 : K=0
        A.y = half ? sql : ly;                   // K=3 : K=1

        // ---- Issue next chunk's load now (uniform path, clamped index) ----
        int mn = (c + 1) * 16 + n;
        int mcn = mn < N_l ? mn : (N_l - 1);
        float nlx = pos_l[3 * mcn + 0];
        float nly = pos_l[3 * mcn + 1];
        float nlz = pos_l[3 * mcn + 2];

        // d2 tile in one matrix op (wave-uniform path, EXEC all ones)
        v8f d = __builtin_amdgcn_wmma_f32_16x16x4_f32(
            false, A, false, B, (short)0, Cinit, false, false);

        const int mbase = c * 16 + half * 8;
#pragma unroll
        for (int j = 0; j < 8; ++j) {
            int mg = mbase + j;
            float d2 = fmaxf(d[j], 0.0f);
            if (mg < N_l && d2 < bd[KNN - 1]) {
                bd[KNN - 1] = d2;
                bi[KNN - 1] = mg;
                // one bubble pass restores sorted order (branchless selects,
                // static indexing only -> stays in VGPRs)
#pragma unroll
                for (int s = KNN - 1; s >= 1; --s) {
                    float a0 = bd[s - 1], a1 = bd[s];
                    int   i0 = bi[s - 1], i1 = bi[s];
                    bool sw = a1 < a0;
                    bd[s - 1] = sw ? a1 : a0;
                    bd[s]     = sw ? a0 : a1;
                    bi[s - 1] = sw ? i1 : i0;
                    bi[s]     = sw ? i0 : i1;
                }
            }
        }

        lx = nlx; ly = nly; lz = nlz;
    }

    // ---- Spill per-lane lists, merge lane pairs (f, f+16) ----
#pragma unroll
    for (int s = 0; s < KNN; ++s) {
        s_ld[tid * KNN + s] = bd[s];
        s_li[tid * KNN + s] = bi[s];
    }
    __syncthreads();

    if (lane < 16) {
        const int ff    = wave * FINE_PER_WAVE + n;   // 0..63
        const int abase = tid * KNN;
        const int bbase = (tid + 16) * KNN;
        int pa = 0, pb = 0;
        float den = 0.f;
        for (int r = 0; r < KNN; ++r) {
            float da = s_ld[abase + pa];
            float db = s_ld[bbase + pb];
            bool ta  = da <= db;
            float d2 = ta ? da : db;
            int   id = ta ? s_li[abase + pa] : s_li[bbase + pb];
            pa += ta ? 1 : 0;
            pb += ta ? 0 : 1;
            float w = 1.0f / fmaxf(d2, 1e-16f);
            den += w;
            s_w [ff * KNN + r] = w;
            s_wi[ff * KNN + r] = id;
        }
        s_den[ff] = den;
    }
    __syncthreads();

    // ---- Weighted feature gather: 8 threads per fine point, 16 ch each ----
    for (int g = 0; g < FINE_PER_BLOCK / 16; ++g) {
        const int ff = g * 16 + (tid >> 3);   // fine point in block
        const int ch = (tid & 7) * 16;        // channel base (64B aligned)
        float4 a0 = {0, 0, 0, 0}, a1 = {0, 0, 0, 0};
        float4 a2 = {0, 0, 0, 0}, a3 = {0, 0, 0, 0};
        for (int k = 0; k < KNN; ++k) {
            float w  = s_w [ff * KNN + k];
            int   id = s_wi[ff * KNN + k];
            const float4* xr = (const float4*)(x + id * FDIM + ch);
            float4 f0 = xr[0], f1 = xr[1], f2 = xr[2], f3 = xr[3];
            a0.x += w * f0.x; a0.y += w * f0.y; a0.z += w * f0.z; a0.w += w * f0.w;
            a1.x += w * f1.x; a1.y += w * f1.y; a1.z += w * f1.z; a1.w += w * f1.w;
            a2.x += w * f2.x; a2.y += w * f2.y; a2.z += w * f2.z; a2.w += w * f2.w;
            a3.x += w * f3.x; a3.y += w * f3.y; a3.z += w * f3.z; a3.w += w * f3.w;
        }
        const int go = fineBase + ff;
        if (go < N_h) {
            float inv = 1.0f / s_den[ff];
            float4* op = (float4*)(out + go * FDIM + ch);
            float4 o0 = {a0.x * inv, a0.y * inv, a0.z * inv, a0.w * inv};
            float4 o1 = {a1.x * inv, a1.y * inv, a1.z * inv, a1.w * inv};
            float4 o2 = {a2.x * inv, a2.y * inv, a2.z * inv, a2.w * inv};
            float4 o3 = {a3.x * inv, a3.y * inv, a3.z * inv, a3.w * inv};
            op[0] = o0; op[1] = o1; op[2] = o2; op[3] = o3;
        }
    }
}

extern "C" void kernel_launch(void* const* d_in, const int* in_sizes, int n_in,
                              void* d_out, int out_size, void* d_ws, size_t ws_size,
                              hipStream_t stream) {
    (void)n_in; (void)out_size; (void)d_ws; (void)ws_size;
    const float* x     = (const float*)d_in[0];   // [N_l, 128]
    const float* pos_l = (const float*)d_in[1];   // [N_l, 3]
    const float* pos_h = (const float*)d_in[2];   // [N_h, 3]
    float*       out   = (float*)d_out;           // [N_h, 128]

    const int N_l = in_sizes[1] / 3;
    const int N_h = in_sizes[2] / 3;

    dim3 grid((N_h + FINE_PER_BLOCK - 1) / FINE_PER_BLOCK);
    dim3 block(BLOCK);
    knn_interp_wmma_kernel<<<grid, block, 0, stream>>>(x, pos_l, pos_h, out, N_l, N_h);
}